// NTMReadHead_44959717654923
// MI455X (gfx1250) — compile-verified
//
#include <hip/hip_runtime.h>
#include <hip/hip_bf16.h>
#include <math.h>

// Problem constants (match reference)
#define BATCH 512
#define EDIM  1024
#define NSLOT 1024
#define MDIM  128
#define PCOLS 134          // M + 6
#define PCOLS_PAD 144      // 9 tiles of 16

typedef __attribute__((ext_vector_type(2))) float v2f;
typedef __attribute__((ext_vector_type(8))) float v8f;

__device__ __forceinline__ float softplus_f(float x) {
    // numerically stable log(1+exp(x))
    if (x > 20.0f) return x;
    if (x < -20.0f) return expf(x);
    return log1pf(expf(x));
}

// ---------------------------------------------------------------------------
// Kernel 0: zero-pad W (1024 x 134) -> Wpad (1024 x 144) so the WMMA GEMM
// inner loop has no bounds checks (no EXEC-mask branches around loads).
// ---------------------------------------------------------------------------
__global__ void __launch_bounds__(256)
pad_W(const float* __restrict__ Wm, float* __restrict__ Wpad) {
    const int idx = blockIdx.x * blockDim.x + threadIdx.x;   // 0 .. 1024*144-1
    if (idx >= EDIM * PCOLS_PAD) return;
    const int row = idx / PCOLS_PAD;
    const int col = idx - row * PCOLS_PAD;
    Wpad[idx] = (col < PCOLS) ? Wm[(size_t)row * PCOLS + col] : 0.0f;
}

// ---------------------------------------------------------------------------
// Kernel 1: params = embeddings @ W + b   via V_WMMA_F32_16X16X4_F32
// Grid: (9 col-tiles, 32 row-tiles), 32 threads (1 wave) per block.
// A-frag (16x4 f32): lanes 0-15 -> M=lane,   VGPR0=K0, VGPR1=K1
//                    lanes16-31 -> M=lane-16,VGPR0=K2, VGPR1=K3
// B-frag (4x16 f32): lane%16 = N; lanes 0-15 -> K0/K1, lanes 16-31 -> K2/K3
// D (16x16 f32): VGPR v: lanes0-15 -> (M=v, N=lane); lanes16-31 -> (M=v+8)
// 4 independent accumulator chains break the WMMA RAW dependency and give
// the scheduler 12 outstanding loads per iteration.
// ---------------------------------------------------------------------------
__global__ void __launch_bounds__(32)
gemm_params_wmma(const float* __restrict__ emb,
                 const float* __restrict__ Wpad,
                 const float* __restrict__ bias,
                 float* __restrict__ params) {
    const int l   = threadIdx.x;            // 0..31
    const int tn  = blockIdx.x;             // 0..8
    const int tm  = blockIdx.y;             // 0..31
    const int row = tm * 16 + (l & 15);     // embeddings row for A
    const int col = tn * 16 + (l & 15);     // Wpad column for B
    const int kh  = (l >> 4) * 2;           // 0 or 2 (K sub-offset per lane half)

    v8f acc0 = {}, acc1 = {}, acc2 = {}, acc3 = {};

    const float* __restrict__ arow = emb + (size_t)row * EDIM;
    const float* __restrict__ bcol = Wpad + col;

    for (int k0 = 0; k0 < EDIM; k0 += 16) {
        {
            const int k = k0 + kh;
            v2f a = *(const v2f*)(arow + k);
            v2f bf;
            bf.x = bcol[(size_t)(k    ) * PCOLS_PAD];
            bf.y = bcol[(size_t)(k + 1) * PCOLS_PAD];
            acc0 = __builtin_amdgcn_wmma_f32_16x16x4_f32(
                false, a, false, bf, (short)0, acc0, false, false);
        }
        {
            const int k = k0 + 4 + kh;
            v2f a = *(const v2f*)(arow + k);
            v2f bf;
            bf.x = bcol[(size_t)(k    ) * PCOLS_PAD];
            bf.y = bcol[(size_t)(k + 1) * PCOLS_PAD];
            acc1 = __builtin_amdgcn_wmma_f32_16x16x4_f32(
                false, a, false, bf, (short)0, acc1, false, false);
        }
        {
            const int k = k0 + 8 + kh;
            v2f a = *(const v2f*)(arow + k);
            v2f bf;
            bf.x = bcol[(size_t)(k    ) * PCOLS_PAD];
            bf.y = bcol[(size_t)(k + 1) * PCOLS_PAD];
            acc2 = __builtin_amdgcn_wmma_f32_16x16x4_f32(
                false, a, false, bf, (short)0, acc2, false, false);
        }
        {
            const int k = k0 + 12 + kh;
            v2f a = *(const v2f*)(arow + k);
            v2f bf;
            bf.x = bcol[(size_t)(k    ) * PCOLS_PAD];
            bf.y = bcol[(size_t)(k + 1) * PCOLS_PAD];
            acc3 = __builtin_amdgcn_wmma_f32_16x16x4_f32(
                false, a, false, bf, (short)0, acc3, false, false);
        }
    }

    if (col < PCOLS) {
        const float bv   = bias[col];
        const int  rbase = tm * 16 + ((l & 16) ? 8 : 0);
        #pragma unroll
        for (int v = 0; v < 8; ++v) {
            params[(size_t)(rbase + v) * PCOLS + col] =
                (acc0[v] + acc1[v]) + (acc2[v] + acc3[v]) + bv;
        }
    }
}

// ---------------------------------------------------------------------------
// Kernel 2: per-batch scalar head params: beta, g, softmax(s), y
// ---------------------------------------------------------------------------
__global__ void __launch_bounds__(256)
head_activations(const float* __restrict__ params, float* __restrict__ stats) {
    const int b = blockIdx.x * blockDim.x + threadIdx.x;
    if (b >= BATCH) return;
    const float* p = params + (size_t)b * PCOLS;
    const float beta = softplus_f(p[MDIM]);
    const float g    = 1.0f / (1.0f + expf(-p[MDIM + 1]));
    const float e0 = p[MDIM + 2], e1 = p[MDIM + 3], e2 = p[MDIM + 4];
    const float mx = fmaxf(e0, fmaxf(e1, e2));
    const float x0 = expf(e0 - mx), x1 = expf(e1 - mx), x2 = expf(e2 - mx);
    const float inv = 1.0f / (x0 + x1 + x2);
    const float y = 1.0f + softplus_f(p[MDIM + 5]);
    float* st = stats + (size_t)b * 8;
    st[0] = beta; st[1] = g;
    st[2] = x0 * inv; st[3] = x1 * inv; st[4] = x2 * inv;
    st[5] = y;
}

// ---------------------------------------------------------------------------
// Kernel 3: fused NTM addressing + read. One workgroup per batch element.
// Pass 1: stream memory[b] once -> dot, ||m||, cos, softmax, gate, shift,
//         sharpen, normalize (all in LDS). Pass 2: stream memory[b] again
//         with w resident in LDS -> memory_data.
// ---------------------------------------------------------------------------
__global__ void __launch_bounds__(256)
ntm_read_main(const float* __restrict__ w_prev,
              const float* __restrict__ mem,
              const float* __restrict__ params,
              const float* __restrict__ stats,
              float* __restrict__ out_md,
              float* __restrict__ out_w) {
    __shared__ __align__(16) float k_s[MDIM];
    __shared__ float arr[NSLOT];   // w_g -> w
    __shared__ float red[256];
    __shared__ float acc2[NSLOT];  // pass-2 partial sums (8 groups x 128 m)

    const int t = threadIdx.x;               // 0..255
    const int b = blockIdx.x;                // 0..511
    const size_t mbase = (size_t)b * NSLOT * MDIM;

    // ---- load k and head scalars ----
    if (t < MDIM) k_s[t] = params[(size_t)b * PCOLS + t];
    const float beta = stats[b * 8 + 0];
    const float g    = stats[b * 8 + 1];
    const float s0   = stats[b * 8 + 2];
    const float s1   = stats[b * 8 + 3];
    const float s2   = stats[b * 8 + 4];
    const float y    = stats[b * 8 + 5];
    __syncthreads();

    // ---- ||k|| ----
    red[t] = (t < MDIM) ? k_s[t] * k_s[t] : 0.0f;
    __syncthreads();
    for (int st_ = 128; st_ > 0; st_ >>= 1) {
        if (t < st_) red[t] += red[t + st_];
        __syncthreads();
    }
    const float k_norm = sqrtf(red[0]);
    __syncthreads();

    // ---- pass 1: dot & row norms over memory[b], 4 rows per thread ----
    const float4* __restrict__ k4 = (const float4*)k_s;
    float xs[4];
    #pragma unroll
    for (int r = 0; r < 4; ++r) {
        const int n = t + r * 256;
        const float4* __restrict__ rowp =
            (const float4*)(mem + mbase + (size_t)n * MDIM);
        float dot = 0.0f, ss = 0.0f;
        #pragma unroll 8
        for (int i = 0; i < MDIM / 4; ++i) {
            const float4 mv = rowp[i];
            const float4 kv = k4[i];
            dot += mv.x * kv.x + mv.y * kv.y + mv.z * kv.z + mv.w * kv.w;
            ss  += mv.x * mv.x + mv.y * mv.y + mv.z * mv.z + mv.w * mv.w;
        }
        const float m_norm = sqrtf(ss);
        const float cosv = dot / (k_norm * m_norm + 1e-16f);
        xs[r] = beta * cosv;
    }

    // ---- softmax over n (max) ----
    red[t] = fmaxf(fmaxf(xs[0], xs[1]), fmaxf(xs[2], xs[3]));
    __syncthreads();
    for (int st_ = 128; st_ > 0; st_ >>= 1) {
        if (t < st_) red[t] = fmaxf(red[t], red[t + st_]);
        __syncthreads();
    }
    const float mx = red[0];
    __syncthreads();

    float ex[4];
    float lsum = 0.0f;
    #pragma unroll
    for (int r = 0; r < 4; ++r) { ex[r] = expf(xs[r] - mx); lsum += ex[r]; }
    red[t] = lsum;
    __syncthreads();
    for (int st_ = 128; st_ > 0; st_ >>= 1) {
        if (t < st_) red[t] += red[t + st_];
        __syncthreads();
    }
    const float invsum = 1.0f / red[0];
    __syncthreads();

    // ---- gate with w_prev -> w_g in LDS ----
    #pragma unroll
    for (int r = 0; r < 4; ++r) {
        const int n = t + r * 256;
        const float w_c = ex[r] * invsum;
        arr[n] = g * w_c + (1.0f - g) * w_prev[(size_t)b * NSLOT + n];
    }
    __syncthreads();

    // ---- circular 3-tap shift (roll +1 / 0 / -1) ----
    float wt[4];
    #pragma unroll
    for (int r = 0; r < 4; ++r) {
        const int n  = t + r * 256;
        const int nm = (n == 0) ? (NSLOT - 1) : n - 1;
        const int np = (n == NSLOT - 1) ? 0 : n + 1;
        wt[r] = s0 * arr[nm] + s1 * arr[n] + s2 * arr[np];
    }

    // ---- sharpen + normalize ----
    float wp[4];
    float psum = 0.0f;
    #pragma unroll
    for (int r = 0; r < 4; ++r) {
        wp[r] = powf(wt[r] + 1e-16f, y);
        psum += wp[r];
    }
    red[t] = psum;
    __syncthreads();   // also separates arr reads (wt) from arr writes below
    for (int st_ = 128; st_ > 0; st_ >>= 1) {
        if (t < st_) red[t] += red[t + st_];
        __syncthreads();
    }
    const float invp = 1.0f / red[0];
    __syncthreads();
    #pragma unroll
    for (int r = 0; r < 4; ++r) {
        const int n = t + r * 256;
        const float w = wp[r] * invp;
        arr[n] = w;
        out_w[(size_t)b * NSLOT + n] = w;
    }
    __syncthreads();

    // ---- pass 2: memory_data = w @ memory[b] ----
    // 8 groups of 32 lanes; each group covers all 128 m via float4/lane,
    // and its own 128-row n-slice -> every load is a coalesced 512B/wave.
    const int grp   = t >> 5;          // 0..7
    const int moff  = (t & 31) * 4;    // float4 start within the row
    const int n0    = grp * (NSLOT / 8);
    float4 acc = make_float4(0.0f, 0.0f, 0.0f, 0.0f);
    #pragma unroll 4
    for (int j = 0; j < NSLOT / 8; ++j) {
        const int n = n0 + j;
        const float wn = arr[n];
        const float4 mv =
            *(const float4*)(mem + mbase + (size_t)n * MDIM + moff);
        acc.x += wn * mv.x;
        acc.y += wn * mv.y;
        acc.z += wn * mv.z;
        acc.w += wn * mv.w;
    }
    acc2[grp * MDIM + moff + 0] = acc.x;
    acc2[grp * MDIM + moff + 1] = acc.y;
    acc2[grp * MDIM + moff + 2] = acc.z;
    acc2[grp * MDIM + moff + 3] = acc.w;
    __syncthreads();
    if (t < MDIM) {
        float sum = 0.0f;
        #pragma unroll
        for (int gg = 0; gg < 8; ++gg) sum += acc2[gg * MDIM + t];
        out_md[(size_t)b * MDIM + t] = sum;
    }
}

// ---------------------------------------------------------------------------
extern "C" void kernel_launch(void* const* d_in, const int* in_sizes, int n_in,
                              void* d_out, int out_size, void* d_ws, size_t ws_size,
                              hipStream_t stream) {
    const float* emb    = (const float*)d_in[0];  // (512, 1024)
    const float* w_prev = (const float*)d_in[1];  // (512, 1024)
    const float* mem    = (const float*)d_in[2];  // (512, 1024, 128)
    const float* Wm     = (const float*)d_in[3];  // (1024, 134)
    const float* bias   = (const float*)d_in[4];  // (134,)

    float* params = (float*)d_ws;                        // 512*134 floats
    float* stats  = params + (size_t)BATCH * PCOLS;      // 512*8 floats
    float* Wpad   = stats + (size_t)BATCH * 8;           // 1024*144 floats

    float* out_md = (float*)d_out;                       // (512, 128)
    float* out_w  = out_md + (size_t)BATCH * MDIM;       // (512, 1024)

    pad_W<<<(EDIM * PCOLS_PAD + 255) / 256, 256, 0, stream>>>(Wm, Wpad);

    dim3 ggrid(PCOLS_PAD / 16, BATCH / 16);              // 9 x 32 tiles
    gemm_params_wmma<<<ggrid, 32, 0, stream>>>(emb, Wpad, bias, params);

    head_activations<<<(BATCH + 255) / 256, 256, 0, stream>>>(params, stats);

    ntm_read_main<<<BATCH, 256, 0, stream>>>(w_prev, mem, params, stats,
                                             out_md, out_w);
}